// MoELayer_41729902248334
// MI455X (gfx1250) — compile-verified
//
#include <hip/hip_runtime.h>
#include <hip/hip_bf16.h>

typedef __attribute__((ext_vector_type(2))) float v2f;
typedef __attribute__((ext_vector_type(8))) float v8f;

constexpr int kN = 4096;
constexpr int kD = 1024;
constexpr int kH = 4096;
constexpr int kE = 8;
constexpr int kHC = 128;   // hidden chunk width per pipeline stage
constexpr int kMT = 32;    // tokens per block tile (two 16-row WMMA halves)

// ---------------------------------------------------------------- zero init
__global__ void moe_zero_kernel(float* __restrict__ out, int* __restrict__ cnt,
                                int total) {
    int i = blockIdx.x * blockDim.x + threadIdx.x;
    if (i < total) out[i] = 0.0f;
    if (i < kE) cnt[i] = 0;
}

// ---------------------------------------------------------------- gating
__global__ __launch_bounds__(256) void moe_gate_kernel(
    const float* __restrict__ x, const float* __restrict__ Wg,
    const float* __restrict__ bg, int* __restrict__ cnt,
    int* __restrict__ tokList, float* __restrict__ wList) {
    int n = blockIdx.x * 256 + threadIdx.x;   // grid sized exactly kN/256
    float acc[kE];
#pragma unroll
    for (int e = 0; e < kE; ++e) acc[e] = bg[e];
    const float* xr = x + (size_t)n * kD;
    for (int k = 0; k < kD; ++k) {
        float xv = xr[k];
        const float* wr = Wg + k * kE;        // uniform address -> scalar loads
#pragma unroll
        for (int e = 0; e < kE; ++e) acc[e] = fmaf(xv, wr[e], acc[e]);
    }
    // softmax over 8 experts
    float m = acc[0];
#pragma unroll
    for (int e = 1; e < kE; ++e) m = fmaxf(m, acc[e]);
    float p[kE]; float ssum = 0.0f;
#pragma unroll
    for (int e = 0; e < kE; ++e) { p[e] = __expf(acc[e] - m); ssum += p[e]; }
    float inv = 1.0f / ssum;
#pragma unroll
    for (int e = 0; e < kE; ++e) p[e] *= inv;
    // top-2 (raw softmax values, not renormalized)
    int i1 = 0; float v1 = p[0];
#pragma unroll
    for (int e = 1; e < kE; ++e) if (p[e] > v1) { v1 = p[e]; i1 = e; }
    int i2 = -1; float v2 = -1.0f;
#pragma unroll
    for (int e = 0; e < kE; ++e)
        if (e != i1 && p[e] > v2) { v2 = p[e]; i2 = e; }

    int p1 = atomicAdd(&cnt[i1], 1);
    tokList[i1 * kN + p1] = n;  wList[i1 * kN + p1] = v1;
    int p2 = atomicAdd(&cnt[i2], 1);
    tokList[i2 * kN + p2] = n;  wList[i2 * kN + p2] = v2;
}

// ---------------------------------------------------------------- fused FFN
// grid.x = kE * (kN/kMT); block = 256 threads (8 waves).
// Block processes 32 gathered tokens of one expert end-to-end; each weight
// (B) fragment is reused by two WMMAs (token rows 0-15 and 16-31).
__global__ __launch_bounds__(256) void moe_ffn_kernel(
    const float* __restrict__ x,  const float* __restrict__ W1,
    const float* __restrict__ b1, const float* __restrict__ W2,
    const float* __restrict__ b2, const int* __restrict__ cnt,
    const int* __restrict__ tokList, const float* __restrict__ wList,
    float* __restrict__ out) {
    __shared__ float sx[kMT][kD + 4];    // x tile, padded for bank spread
    __shared__ float sh[kMT][kHC + 4];   // relu(h) chunk
    __shared__ int   stok[kMT];
    __shared__ float swgt[kMT];

    const int e    = blockIdx.x >> 7;        // kN/kMT = 128 tiles
    const int tile = blockIdx.x & 127;
    const int cnt_e = cnt[e];
    if (tile * kMT >= cnt_e) return;         // uniform per block

    const int lane   = threadIdx.x & 31;
    const int laneLo = lane & 15;
    const int laneHi = lane >> 4;
    const int wv     = threadIdx.x >> 5;     // wave id 0..7

    if (threadIdx.x < kMT) {
        int r = tile * kMT + threadIdx.x;
        if (r < cnt_e) {
            stok[threadIdx.x] = tokList[e * kN + r];
            swgt[threadIdx.x] = wList[e * kN + r];
        } else {
            stok[threadIdx.x] = 0;
            swgt[threadIdx.x] = 0.0f;
        }
    }
    __syncthreads();

    // stage 32 x-rows (gathered) into LDS, float4 vectorized
#pragma unroll
    for (int j = 0; j < 32; ++j) {
        int idx = threadIdx.x + j * 256;     // 0..8191 over float4 space
        int row = idx >> 8;
        int c4  = idx & 255;
        const float4 xv =
            ((const float4*)(x + (size_t)stok[row] * kD))[c4];
        float* dst = &sx[row][c4 * 4];
        dst[0] = xv.x; dst[1] = xv.y; dst[2] = xv.z; dst[3] = xv.w;
    }

    // output accumulators: wave owns cols [wv*128, +128); [rowHalf][subtile]
    v8f oacc[2][8];
#pragma unroll
    for (int s = 0; s < 8; ++s) {
        float bv = b2[(size_t)e * kD + wv * 128 + s * 16 + laneLo];
#pragma unroll
        for (int j = 0; j < 8; ++j) { oacc[0][s][j] = bv; oacc[1][s][j] = bv; }
    }
    __syncthreads();

    for (int hc = 0; hc < kH; hc += kHC) {
        // ---- phase 1: wave computes h subtiles (32 rows) for cols [hc+wv*16, +16)
        const int hcol = hc + wv * 16 + laneLo;
        v8f hacc0, hacc1;
        {
            float b1v = b1[(size_t)e * kH + hcol];
#pragma unroll
            for (int j = 0; j < 8; ++j) { hacc0[j] = b1v; hacc1[j] = b1v; }
        }
        const float* W1p = W1 + (size_t)e * kD * kH + hcol;  // W1[e][k][hcol]
#pragma unroll 4
        for (int k = 0; k < kD; k += 4) {
            int kk = k + 2 * laneHi;
            v2f a0 = *(const v2f*)&sx[laneLo][kk];
            v2f a1 = *(const v2f*)&sx[laneLo + 16][kk];
            v2f b;
            b[0] = W1p[(size_t)kk * kH];
            b[1] = W1p[(size_t)(kk + 1) * kH];
            hacc0 = __builtin_amdgcn_wmma_f32_16x16x4_f32(
                false, a0, false, b, (short)0, hacc0, false, false);
            hacc1 = __builtin_amdgcn_wmma_f32_16x16x4_f32(
                false, a1, false, b, (short)0, hacc1, false, false);
        }
        __syncthreads();   // previous chunk's consumers are done with sh
#pragma unroll
        for (int j = 0; j < 8; ++j) {
            sh[j + 8 * laneHi][wv * 16 + laneLo]      = fmaxf(hacc0[j], 0.0f);
            sh[16 + j + 8 * laneHi][wv * 16 + laneLo] = fmaxf(hacc1[j], 0.0f);
        }
        __syncthreads();   // sh chunk ready

        // ---- phase 2: out += sh(32xkHC) @ W2[e][hc:hc+kHC, cols]
#pragma unroll
        for (int s = 0; s < 8; ++s) {
            const int col = wv * 128 + s * 16 + laneLo;
            const float* W2p = W2 + ((size_t)e * kH + hc) * kD + col;
            v8f acc0 = oacc[0][s];
            v8f acc1 = oacc[1][s];
#pragma unroll 4
            for (int k2 = 0; k2 < kHC; k2 += 4) {
                int kk = k2 + 2 * laneHi;
                v2f a0 = *(const v2f*)&sh[laneLo][kk];
                v2f a1 = *(const v2f*)&sh[laneLo + 16][kk];
                v2f b;
                b[0] = W2p[(size_t)kk * kD];
                b[1] = W2p[(size_t)(kk + 1) * kD];
                acc0 = __builtin_amdgcn_wmma_f32_16x16x4_f32(
                    false, a0, false, b, (short)0, acc0, false, false);
                acc1 = __builtin_amdgcn_wmma_f32_16x16x4_f32(
                    false, a1, false, b, (short)0, acc1, false, false);
            }
            oacc[0][s] = acc0;
            oacc[1][s] = acc1;
        }
    }

    // ---- scatter: out[token] += w * (y + b2)   (b2 already in oacc)
#pragma unroll
    for (int s = 0; s < 8; ++s) {
        const int col = wv * 128 + s * 16 + laneLo;
#pragma unroll
        for (int j = 0; j < 8; ++j) {
            int row0 = j + 8 * laneHi;
            int row1 = row0 + 16;
            atomicAdd(&out[(size_t)stok[row0] * kD + col],
                      swgt[row0] * oacc[0][s][j]);
            atomicAdd(&out[(size_t)stok[row1] * kD + col],
                      swgt[row1] * oacc[1][s][j]);
        }
    }
}

// ---------------------------------------------------------------- launcher
extern "C" void kernel_launch(void* const* d_in, const int* in_sizes, int n_in,
                              void* d_out, int out_size, void* d_ws, size_t ws_size,
                              hipStream_t stream) {
    const float* x  = (const float*)d_in[0];
    const float* Wg = (const float*)d_in[1];
    const float* bg = (const float*)d_in[2];
    const float* W1 = (const float*)d_in[3];
    const float* b1 = (const float*)d_in[4];
    const float* W2 = (const float*)d_in[5];
    const float* b2 = (const float*)d_in[6];
    float* out = (float*)d_out;

    // workspace layout
    char* ws = (char*)d_ws;
    int*   cnt     = (int*)ws;                                // 8 ints
    int*   tokList = (int*)(ws + 64);                         // kE*kN ints
    float* wList   = (float*)(ws + 64 + (size_t)kE * kN * 4); // kE*kN floats

    int total = kN * kD;
    moe_zero_kernel<<<(total + 255) / 256, 256, 0, stream>>>(out, cnt, total);
    moe_gate_kernel<<<kN / 256, 256, 0, stream>>>(x, Wg, bg, cnt, tokList, wList);
    moe_ffn_kernel<<<kE * (kN / kMT), 256, 0, stream>>>(
        x, W1, b1, W2, b2, cnt, tokList, wList, out);
}